// ROIAlign_38534446579979
// MI455X (gfx1250) — compile-verified
//
#include <hip/hip_runtime.h>

// ROI Align on MI455X (gfx1250, wave32).
// B=8, H=W=32, C=256, R=128 ROIs/img, OUT=7, SAMPLE_RATIO=2 -> S=14.
//
// Design: one workgroup (8 wave32) per ROI.
//   stage 1 (VALU):  y-interp of two feature rows -> LDS in WMMA-B-fragment
//                    interleaved layout (x-pairs interleaved per channel),
//                    double-buffered 2 x 32 KB.
//   stage 2 (WMMA):  fused x-interp + 2x2 pooling as GEMM
//                    out[oy] += P(16x32) x rowint(32x256), K in steps of 4 via
//                    v_wmma_f32_16x16x4_f32; B fragments are single ds_load_b64s.
// Roofline: output writes 51.4 MB dominate HBM (features are L2-resident in the
// 192 MB L2) -> ~2.2 us floor at 23.3 TB/s; WMMA work is 3.8 GFLOP f32, far
// below any compute ceiling, so matrix pipes are free and VALU handles lerp.

#define OUTSZ 7
#define S     14
#define NROI  128
#define BB    8
#define HH    32
#define WW    32
#define CC    256

typedef __attribute__((ext_vector_type(2))) float v2f;
typedef __attribute__((ext_vector_type(8))) float v8f;

__global__ __launch_bounds__(256) void roi_align_wmma(
    const float* __restrict__ fmap,   // [8,32,32,256]
    const float* __restrict__ boxes,  // [8,128,4]
    float* __restrict__ out)          // [8,128,7,7,256]
{
    __shared__ float s_P[16][32];            // pooled x-weight matrix (GEMM A), 2 KB
    __shared__ int   s_y0[S], s_y1[S];
    __shared__ float s_wy[S];
    __shared__ int   s_x0[S], s_x1[S];
    __shared__ float s_wx[S];
    // Interleaved y-interp buffer: element (x, c) at dword (x>>1)*512 + c*2 + (x&1).
    __shared__ __align__(16) float s_row[2][WW * CC];   // 2 x 32 KB

    const int roi  = blockIdx.x;             // 0..1023
    const int b    = roi >> 7;               // roi / NROI
    const int tid  = threadIdx.x;
    const int lane = tid & 31;
    const int wave = tid >> 5;               // 0..7  -> channel block of 32

    // ---------------- per-ROI sampling metadata (matches reference math) ---
    const float inv31 = 1.0f / 31.0f;
    const float bx0 = boxes[roi * 4 + 0] * inv31;
    const float by0 = boxes[roi * 4 + 1] * inv31;
    const float bx2 = boxes[roi * 4 + 2] * inv31;
    const float by2 = boxes[roi * 4 + 3] * inv31;
    const float x1 = fmaxf(bx0, 0.0f), y1 = fmaxf(by0, 0.0f);
    const float x2 = fminf(bx2, 1.0f), y2 = fminf(by2, 1.0f);
    const float bin_h = (y2 - y1) * (1.0f / OUTSZ);
    const float bin_w = (x2 - x1) * (1.0f / OUTSZ);
    const float gy1 = y1 + 0.25f * bin_h, gy2 = y2 - 0.25f * bin_h;
    const float gx1 = x1 + 0.25f * bin_w, gx2 = x2 - 0.25f * bin_w;
    const float ystep = (gy2 - gy1) * 31.0f / (float)(S - 1);
    const float xstep = (gx2 - gx1) * 31.0f / (float)(S - 1);

    if (tid < S) {
        float ys = gy1 * 31.0f + (float)tid * ystep;
        float yf = floorf(ys);
        int   yi = (int)yf;
        s_wy[tid] = ys - yf;
        s_y0[tid] = min(max(yi, 0), HH - 1);
        s_y1[tid] = min(max(yi + 1, 0), HH - 1);
        float xs = gx1 * 31.0f + (float)tid * xstep;
        float xf = floorf(xs);
        int   xi = (int)xf;
        s_wx[tid] = xs - xf;
        s_x0[tid] = min(max(xi, 0), WW - 1);
        s_x1[tid] = min(max(xi + 1, 0), WW - 1);
    }
    __syncthreads();

    // P[ox][x] = 1/4 * sum_{sx in {2ox,2ox+1}} x-bilinear-weight(sx, x); rows 7..15 zero.
    for (int e = tid; e < 16 * 32; e += 256) {
        int ox = e >> 5, x = e & 31;
        float p = 0.0f;
        if (ox < OUTSZ) {
            #pragma unroll
            for (int j = 0; j < 2; ++j) {
                int sx = 2 * ox + j;
                float w = 0.0f;
                if (x == s_x0[sx]) w += 1.0f - s_wx[sx];
                if (x == s_x1[sx]) w += s_wx[sx];
                p += w;
            }
            p *= 0.25f;
        }
        s_P[ox][x] = p;
    }
    __syncthreads();

    // A fragments, 16x4 f32 layout: VGPR0 holds K (lanes 0-15 -> K+0, 16-31 -> K+2),
    //                               VGPR1 holds K+1 / K+3.
    v2f afrag[8];
    {
        const int m     = lane & 15;
        const int khalf = (lane >> 4) << 1;  // 0 or 2
        #pragma unroll
        for (int kk = 0; kk < 8; ++kk) {
            afrag[kk].x = s_P[m][4 * kk + khalf + 0];
            afrag[kk].y = s_P[m][4 * kk + khalf + 1];
        }
    }

    const float* fbase  = fmap + (size_t)b * HH * WW * CC;
    const int    cbase  = wave * 32;           // this wave's 32 channels
    const int    nc     = cbase + (lane & 15); // N index within tile 0
    const int    prbase = lane >> 4;           // row-pair half select (0 or 1)
    const int    mbase  = (lane >> 4) << 3;    // C/D: lanes>=16 hold M+8

    for (int oy = 0; oy < OUTSZ; ++oy) {
        v8f acc0 = {};
        v8f acc1 = {};
        #pragma unroll
        for (int j = 0; j < 2; ++j) {
            const int    sy  = 2 * oy + j;
            float*       buf = s_row[sy & 1];
            const float* r0  = fbase + (size_t)s_y0[sy] * (WW * CC);
            const float* r1  = fbase + (size_t)s_y1[sy] * (WW * CC);
            const float  wy  = s_wy[sy];

            // stage 1: y-interp 32x256 -> LDS in interleaved B-fragment layout.
            // Each thread handles one (x-pair, channel-pair) 2x2 patch.
            const float2* r0v = (const float2*)r0;   // index = x*128 + cp
            const float2* r1v = (const float2*)r1;
            float4*       bv  = (float4*)buf;        // index = a*128 + cp
            #pragma unroll
            for (int k = 0; k < 8; ++k) {
                const int e  = tid + k * 256;
                const int a  = e >> 7;       // x pair index (x = 2a, 2a+1)
                const int cp = e & 127;      // channel pair index (c = 2cp)
                const int i0 = (2 * a) * 128 + cp;
                float2 g00 = r0v[i0],       g01 = r0v[i0 + 128];
                float2 g10 = r1v[i0],       g11 = r1v[i0 + 128];
                float4 o;                    // {(2a,c),(2a+1,c),(2a,c+1),(2a+1,c+1)}
                o.x = g00.x + wy * (g10.x - g00.x);
                o.y = g01.x + wy * (g11.x - g01.x);
                o.z = g00.y + wy * (g10.y - g00.y);
                o.w = g01.y + wy * (g11.y - g01.y);
                bv[a * 128 + cp] = o;
            }
            // prefetch next sample row's feature rows into the cache hierarchy
            if (sy + 1 < S) {
                const float* n0 = fbase + (size_t)s_y0[sy + 1] * (WW * CC);
                const float* n1 = fbase + (size_t)s_y1[sy + 1] * (WW * CC);
                __builtin_prefetch(n0 + tid * 32, 0, 3);
                __builtin_prefetch(n1 + tid * 32, 0, 3);
            }
            __syncthreads();  // buf[sy&1] ready (double buffer -> one barrier/row)

            // stage 2: acc += P(16x32) x buf(32 x 16) per tile, K in steps of 4.
            // B fragment = one contiguous b64: {row 4kk+2h, row 4kk+2h+1} at col n.
            #pragma unroll
            for (int kk = 0; kk < 8; ++kk) {
                const int base = (2 * kk + prbase) * 512 + nc * 2;
                const v2f bf0 = *(const v2f*)(buf + base);
                const v2f bf1 = *(const v2f*)(buf + base + 32);
                acc0 = __builtin_amdgcn_wmma_f32_16x16x4_f32(
                        false, afrag[kk], false, bf0, (short)0, acc0, false, false);
                acc1 = __builtin_amdgcn_wmma_f32_16x16x4_f32(
                        false, afrag[kk], false, bf1, (short)0, acc1, false, false);
            }
        }

        // store output row oy: D layout -> lane L, VGPR v: M = v + (L<16?0:8), N = L%16
        float* orow = out + ((size_t)roi * (OUTSZ * OUTSZ) + (size_t)oy * OUTSZ) * CC;
        #pragma unroll
        for (int v = 0; v < 8; ++v) {
            int m = mbase + v;
            if (m < OUTSZ) {
                orow[(size_t)m * CC + nc]      = acc0[v];
                orow[(size_t)m * CC + nc + 16] = acc1[v];
            }
        }
    }
}

extern "C" void kernel_launch(void* const* d_in, const int* in_sizes, int n_in,
                              void* d_out, int out_size, void* d_ws, size_t ws_size,
                              hipStream_t stream) {
    const float* fmap  = (const float*)d_in[0];
    const float* boxes = (const float*)d_in[1];
    float*       out   = (float*)d_out;
    roi_align_wmma<<<dim3(BB * NROI), dim3(256), 0, stream>>>(fmap, boxes, out);
}